// Policy_79190607003709
// MI455X (gfx1250) — compile-verified
//
#include <hip/hip_runtime.h>

// Problem constants (reference: B=65536, A=8, G=25, F=3; out feature = 6)
#define GSZ 25
#define LUT_NODES 50     // G*2 coords
#define LUT_ADJ   625    // G*G adjacency
#define LUT_TOTAL 675    // dwords staged to LDS

typedef __attribute__((ext_vector_type(4))) float v4f;

typedef __attribute__((address_space(1))) int* gptr_i32;
typedef __attribute__((address_space(3))) int* lptr_i32;

__global__ __launch_bounds__(256) void gather_concat_kernel(
    const float* __restrict__ state,   // [B, G, 3]
    const int*   __restrict__ nlv,     // [B, A] = [B*A]
    const float* __restrict__ nodes,   // [G, 2]
    const float* __restrict__ adj,     // [G, G]
    float* __restrict__ out,           // [B, A, G, 6]
    unsigned ncells)                   // B*A*G
{
    __shared__ float lut[LUT_TOTAL];   // [0,50) = nodes, [50,675) = adj

    // ---- Stage tiny LUTs into LDS via gfx1250 async global->LDS path ----
#if __has_builtin(__builtin_amdgcn_global_load_async_to_lds_b32)
    for (unsigned i = threadIdx.x; i < LUT_TOTAL; i += blockDim.x) {
        const float* gsrc = (i < LUT_NODES) ? (nodes + i) : (adj + (i - LUT_NODES));
        __builtin_amdgcn_global_load_async_to_lds_b32(
            (gptr_i32)const_cast<float*>(gsrc),
            (lptr_i32)&lut[i],
            /*offset=*/0, /*cpol=*/0);
    }
#if __has_builtin(__builtin_amdgcn_s_wait_asynccnt)
    __builtin_amdgcn_s_wait_asynccnt(0);
#else
    asm volatile("s_wait_asynccnt 0" ::: "memory");
#endif
#else
    for (unsigned i = threadIdx.x; i < LUT_TOTAL; i += blockDim.x)
        lut[i] = (i < LUT_NODES) ? nodes[i] : adj[i - LUT_NODES];
#endif
    __syncthreads();

    // ---- Each thread builds 2 cells (12 floats) -> 3 x b128 NT stores ----
    unsigned t  = blockIdx.x * blockDim.x + threadIdx.x;
    unsigned c0 = t * 2u;
    if (c0 >= ncells) return;          // ncells is even; c0+1 is in range

    unsigned ba0 = c0 / GSZ;           // (b*A + a), magic-mul division
    unsigned g0  = c0 - ba0 * GSZ;
    unsigned g1  = g0 + 1u;
    unsigned ba1 = ba0;
    if (g1 == GSZ) { g1 = 0u; ba1 = ba0 + 1u; }

    int idx0 = nlv[ba0];
    int idx1 = (ba1 == ba0) ? idx0 : nlv[ba1];

    unsigned b0 = ba0 >> 3;            // A == 8
    unsigned b1 = ba1 >> 3;

    const float* s0 = state + (b0 * (GSZ * 3u) + g0 * 3u);
    const float* s1 = state + (b1 * (GSZ * 3u) + g1 * 3u);
    float s00 = s0[0], s01 = s0[1], s02 = s0[2];
    float s10 = s1[0], s11 = s1[1], s12 = s1[2];

    float n00 = lut[idx0 * 2],     n01 = lut[idx0 * 2 + 1];
    float n10 = lut[idx1 * 2],     n11 = lut[idx1 * 2 + 1];
    float m0  = lut[LUT_NODES + idx0 * GSZ + (int)g0];
    float m1  = lut[LUT_NODES + idx1 * GSZ + (int)g1];

    v4f o0 = {s00, s01, s02, n00};     // cell0: state012, node_x
    v4f o1 = {n01, m0,  s10, s11};     // cell0: node_y, mask | cell1: state01
    v4f o2 = {s12, n10, n11, m1};      // cell1: state2, node_xy, mask

    v4f* op = (v4f*)out + (size_t)t * 3u;
    __builtin_nontemporal_store(o0, op + 0);
    __builtin_nontemporal_store(o1, op + 1);
    __builtin_nontemporal_store(o2, op + 2);
}

extern "C" void kernel_launch(void* const* d_in, const int* in_sizes, int n_in,
                              void* d_out, int out_size, void* d_ws, size_t ws_size,
                              hipStream_t stream) {
    const float* state = (const float*)d_in[0];   // [B,G,3]
    const int*   nlv   = (const int*)d_in[1];     // [B,A]
    const float* nodes = (const float*)d_in[2];   // [G,2]
    const float* adj   = (const float*)d_in[3];   // [G,G]
    float*       out   = (float*)d_out;           // [B,A,G,6]

    unsigned ncells   = (unsigned)(out_size / 6); // B*A*G = 13,107,200
    unsigned nthreads = ncells / 2u;              // 2 cells per thread
    dim3 block(256);
    dim3 grid((nthreads + block.x - 1) / block.x);
    gather_concat_kernel<<<grid, block, 0, stream>>>(state, nlv, nodes, adj, out, ncells);
}